// Model_8495445311643
// MI455X (gfx1250) — compile-verified
//
#include <hip/hip_runtime.h>
#include <hip/hip_bf16.h>
#include <math.h>

// Problem constants (from reference)
#define BB    2
#define NN    64
#define TT    96
#define DM    128
#define BT    (BB*TT)
#define HID   32
#define EDIM  4

// LDS leading dims (pad so 16-lane column reads hit distinct banks: ld%64==4)
#define Z_LD  132
#define L_LD  68
#define QH_LD 33

typedef __attribute__((ext_vector_type(2))) float v2f;
typedef __attribute__((ext_vector_type(8))) float v8f;

// ---- WMMA f32 16x16x4 fragment helpers (wave32 layouts per CDNA5 ISA 7.12.2) ----
// A 16x4 f32: lane L -> m = L&15, VGPR0 = A[m][k + 2*(L>>4)], VGPR1 = A[m][k + 2*(L>>4) + 1]
__device__ __forceinline__ v2f frag_a(const float* p, int ld, int m0, int k, int lane) {
  int m  = m0 + (lane & 15);
  int kk = k + ((lane >> 4) << 1);
  v2f a; a.x = p[m * ld + kk]; a.y = p[m * ld + kk + 1]; return a;
}
// B 4x16 f32 from row-major B[k][n]
__device__ __forceinline__ v2f frag_b(const float* p, int ld, int k, int n0, int lane) {
  int n  = n0 + (lane & 15);
  int kk = k + ((lane >> 4) << 1);
  v2f b; b.x = p[kk * ld + n]; b.y = p[(kk + 1) * ld + n]; return b;
}
// B 4x16 f32 where the logical B is the transpose of a row-major matrix M: B[k][n] = M[n][k]
__device__ __forceinline__ v2f frag_bt(const float* p, int ld, int k, int n0, int lane) {
  int n  = n0 + (lane & 15);
  int kk = k + ((lane >> 4) << 1);
  v2f b; b.x = p[n * ld + kk]; b.y = p[n * ld + kk + 1]; return b;
}

__device__ __forceinline__ v8f wmma4(v2f a, v2f b, v8f c) {
  return __builtin_amdgcn_wmma_f32_16x16x4_f32(false, a, false, b, (short)0, c,
                                               false, false);
}

// ---- CDNA5 async copy: global -> LDS, 16 bytes per lane, tracked by ASYNCcnt ----
__device__ __forceinline__ void async_b128(void* lds_ptr, const void* gptr) {
  unsigned lds_addr = (unsigned)(unsigned long long)lds_ptr; // low 32b of generic = LDS offset
  unsigned long long ga = (unsigned long long)gptr;
  asm volatile("global_load_async_to_lds_b128 %0, %1, off"
               :: "v"(lds_addr), "v"(ga) : "memory");
}
__device__ __forceinline__ void wait_async0() {
  asm volatile("s_wait_asynccnt 0" ::: "memory");
}

__global__ __launch_bounds__(256) void fused_spatial_attn(
    const float* __restrict__ x,        // [B,N,T,D]
    const float* __restrict__ ef,       // [BT,N,N,4]
    const float* __restrict__ ap,       // [BT,N,N,5]
    const unsigned char* __restrict__ padmask, // [B,N] bool
    const float* __restrict__ Wq, const float* __restrict__ Wk,
    const float* __restrict__ Wv,
    const float* __restrict__ W1,       // [260,32]
    const float* __restrict__ b1, const float* __restrict__ W2,
    const float* __restrict__ b2, const float* __restrict__ Wfuse,
    const float* __restrict__ Wtheta,
    const float* __restrict__ lng, const float* __restrict__ lnb,
    const float* __restrict__ physw_p, const float* __restrict__ priorw_p,
    float* __restrict__ out)            // [B,N,T,D]
{
  __shared__ __align__(16) float Zs[NN * Z_LD];  // Z slab (kept for residual)
  __shared__ __align__(16) float Ws[DM * DM];    // staged weight matrix (Wq/Wk/Wv/Wtheta)
  __shared__ float Qs[NN * Z_LD];   // Q, later reused for `spatial`
  __shared__ float Ks[NN * Z_LD];   // K, later reused for h = Z + out_feat
  __shared__ float Vs[NN * Z_LD];   // V
  __shared__ float Ls[NN * L_LD];   // content logits -> logits -> alpha
  __shared__ float qhs[NN * QH_LD]; // Q @ W1[:D]
  __shared__ float khs[NN * QH_LD]; // K @ W1[D:2D]
  __shared__ float b1s[HID], W2s[HID], W1es[EDIM * HID];

  const int bt   = blockIdx.x;
  const int b    = bt / TT;
  const int t    = bt % TT;
  const int tid  = threadIdx.x;
  const int lane = tid & 31;
  const int wave = tid >> 5;
  const unsigned char* maskb = padmask + b * NN;

  // Warm L2 for the weights we will DMA later (global_prefetch_b8).
  __builtin_prefetch(Wk + tid * 64, 0, 2);
  __builtin_prefetch(Wv + tid * 64, 0, 2);
  __builtin_prefetch(Wtheta + tid * 64, 0, 2);

  // ---- Phase 1: async-stage Z[bt] = x[b,:,t,:] (64x128) and Wq into LDS ----
  for (int i = tid; i < NN * (DM / 4); i += 256) {   // 16B chunks of Z
    int n  = i >> 5;
    int d4 = i & 31;
    async_b128(&Zs[n * Z_LD + d4 * 4],
               x + ((size_t)(b * NN + n) * TT + t) * DM + d4 * 4);
  }
  for (int i = tid; i < DM * DM / 4; i += 256)       // 16B chunks of Wq
    async_b128(&Ws[i * 4], Wq + i * 4);
  if (tid < HID)        { b1s[tid] = b1[tid]; W2s[tid] = W2[tid]; }
  if (tid < EDIM * HID) { W1es[tid] = W1[2 * DM * HID + tid]; }   // W1[256+c][h]
  wait_async0();
  __syncthreads();

  // ---- Phase 2: Q/K/V = Z @ Wq/Wk/Wv, weight staged in LDS, pure ds+wmma loop ----
  for (int g = 0; g < 3; ++g) {
    float* D = (g == 0) ? Qs : ((g == 1) ? Ks : Vs);
    for (int tile = wave; tile < 32; tile += 8) {
      int m0 = (tile >> 3) << 4;
      int n0 = (tile & 7) << 4;
      v8f acc = {};
      for (int k = 0; k < DM; k += 4)
        acc = wmma4(frag_a(Zs, Z_LD, m0, k, lane), frag_b(Ws, DM, k, n0, lane), acc);
      int col = n0 + (lane & 15);
      int rb  = m0 + ((lane >> 4) << 3);
#pragma unroll
      for (int r = 0; r < 8; ++r) D[(rb + r) * Z_LD + col] = acc[r];
    }
    __syncthreads();                       // everyone done reading Ws
    if (g < 2) {
      const float* Wn = (g == 0) ? Wk : Wv;
      for (int i = tid; i < DM * DM / 4; i += 256)
        async_b128(&Ws[i * 4], Wn + i * 4);
      wait_async0();
      __syncthreads();
    }
  }

  // Kick off the Wtheta DMA now; it overlaps phases 3-5 and completes before 6b.
  for (int i = tid; i < DM * DM / 4; i += 256)
    async_b128(&Ws[i * 4], Wtheta + i * 4);

  // ---- Phase 3: content logits Q@K^T (16 tiles), qh=Q@W1a (8), kh=K@W1b (8) ----
  for (int job = wave; job < 32; job += 8) {
    if (job < 16) {                    // content logits, raw dot products
      int m0 = (job >> 2) << 4;
      int n0 = (job & 3) << 4;
      v8f acc = {};
      for (int k = 0; k < DM; k += 4)
        acc = wmma4(frag_a(Qs, Z_LD, m0, k, lane), frag_bt(Ks, Z_LD, k, n0, lane), acc);
      int col = n0 + (lane & 15);
      int rb  = m0 + ((lane >> 4) << 3);
#pragma unroll
      for (int r = 0; r < 8; ++r) Ls[(rb + r) * L_LD + col] = acc[r];
    } else {                           // qh / kh
      int q   = job - 16;              // 0..15
      int kh  = q >> 3;                // 0: qh, 1: kh
      int tt2 = q & 7;                 // 4x2 tiles
      int m0  = (tt2 >> 1) << 4;
      int n0  = (tt2 & 1) << 4;
      const float* A  = kh ? Ks : Qs;
      const float* Bw = W1 + (kh ? DM * HID : 0);
      float* D = kh ? khs : qhs;
      v8f acc = {};
      for (int k = 0; k < DM; k += 4)
        acc = wmma4(frag_a(A, Z_LD, m0, k, lane), frag_b(Bw, HID, k, n0, lane), acc);
      int col = n0 + (lane & 15);
      int rb  = m0 + ((lane >> 4) << 3);
#pragma unroll
      for (int r = 0; r < 8; ++r) D[(rb + r) * QH_LD + col] = acc[r];
    }
  }
  __syncthreads();

  // ---- Phase 4: fused edge-MLP + prior + mask -> complete logits ----
  {
    const float physw  = physw_p[0];
    const float priorw = priorw_p[0];
    const float b2v    = b2[0];
    const float wf0 = Wfuse[0], wf1 = Wfuse[1], wf2 = Wfuse[2], wf3 = Wfuse[3],
                wf4 = Wfuse[4];
    const float invs = rsqrtf((float)DM);
    for (int p = tid; p < NN * NN; p += 256) {
      int i = p >> 6, j = p & 63;
      const float* e = ef + ((size_t)bt * NN * NN + p) * EDIM;
      float e0 = e[0], e1 = e[1], e2 = e[2], e3 = e[3];
      float acc = 0.f;
#pragma unroll
      for (int h = 0; h < HID; ++h) {
        float hv = qhs[i * QH_LD + h] + khs[j * QH_LD + h] + b1s[h]
                 + e0 * W1es[h] + e1 * W1es[HID + h]
                 + e2 * W1es[2 * HID + h] + e3 * W1es[3 * HID + h];
        acc += fmaxf(hv, 0.f) * W2s[h];
      }
      float phys = acc + b2v;
      const float* a5 = ap + ((size_t)bt * NN * NN + p) * 5;
      float fa = a5[0] * wf0 + a5[1] * wf1 + a5[2] * wf2 + a5[3] * wf3 + a5[4] * wf4;
      if (!isfinite(fa)) fa = 0.f;           // nan_to_num(nan/±inf -> 0)
      fa = fmaxf(fa, 0.f);
      float lg = Ls[i * L_LD + j] * invs + physw * phys + priorw * __logf(fa + 1e-6f);
      if (maskb[i] || maskb[j]) lg = -1e9f;
      Ls[i * L_LD + j] = lg;
    }
  }
  __syncthreads();

  // ---- Phase 5: softmax over j (wave per row, 2 cols per lane) ----
  for (int row = wave; row < NN; row += 8) {
    float l0 = Ls[row * L_LD + lane];
    float l1 = Ls[row * L_LD + lane + 32];
    float m = fmaxf(l0, l1);
    for (int off = 16; off; off >>= 1) m = fmaxf(m, __shfl_xor(m, off, 32));
    float x0 = __expf(l0 - m), x1 = __expf(l1 - m);
    float s = x0 + x1;
    for (int off = 16; off; off >>= 1) s += __shfl_xor(s, off, 32);
    float inv = 1.f / s;
    Ls[row * L_LD + lane]      = x0 * inv;
    Ls[row * L_LD + lane + 32] = x1 * inv;
  }
  __syncthreads();

  // ---- Phase 6a: spatial = alpha @ V (reuse Qs) ----
  for (int job = wave; job < 32; job += 8) {
    int m0 = (job >> 3) << 4;
    int n0 = (job & 7) << 4;
    v8f acc = {};
    for (int k = 0; k < NN; k += 4)
      acc = wmma4(frag_a(Ls, L_LD, m0, k, lane), frag_b(Vs, Z_LD, k, n0, lane), acc);
    int col = n0 + (lane & 15);
    int rb  = m0 + ((lane >> 4) << 3);
#pragma unroll
    for (int r = 0; r < 8; ++r) Qs[(rb + r) * Z_LD + col] = acc[r];
  }
  wait_async0();                       // Wtheta DMA (issued before phase 3) has landed
  __syncthreads();

  // ---- Phase 6b: h = Z + spatial @ Wtheta (Wtheta in Ws; reuse Ks for h) ----
  for (int job = wave; job < 32; job += 8) {
    int m0 = (job >> 3) << 4;
    int n0 = (job & 7) << 4;
    v8f acc = {};
    for (int k = 0; k < DM; k += 4)
      acc = wmma4(frag_a(Qs, Z_LD, m0, k, lane), frag_b(Ws, DM, k, n0, lane), acc);
    int col = n0 + (lane & 15);
    int rb  = m0 + ((lane >> 4) << 3);
#pragma unroll
    for (int r = 0; r < 8; ++r)
      Ks[(rb + r) * Z_LD + col] = Zs[(rb + r) * Z_LD + col] + acc[r];
  }
  __syncthreads();

  // ---- Phase 7: LayerNorm over D + masked coalesced store to [B,N,T,D] ----
  for (int row = wave; row < NN; row += 8) {
    float v0 = Ks[row * Z_LD + lane];
    float v1 = Ks[row * Z_LD + lane + 32];
    float v2 = Ks[row * Z_LD + lane + 64];
    float v3 = Ks[row * Z_LD + lane + 96];
    float s = v0 + v1 + v2 + v3;
    float q = v0 * v0 + v1 * v1 + v2 * v2 + v3 * v3;
    for (int off = 16; off; off >>= 1) {
      s += __shfl_xor(s, off, 32);
      q += __shfl_xor(q, off, 32);
    }
    float mu  = s * (1.f / DM);
    float var = q * (1.f / DM) - mu * mu;
    float rs  = rsqrtf(var + 1e-5f);
    float keep = maskb[row] ? 0.f : 1.f;
    float* o = out + ((size_t)(b * NN + row) * TT + t) * DM;
    o[lane]      = ((v0 - mu) * rs * lng[lane]      + lnb[lane])      * keep;
    o[lane + 32] = ((v1 - mu) * rs * lng[lane + 32] + lnb[lane + 32]) * keep;
    o[lane + 64] = ((v2 - mu) * rs * lng[lane + 64] + lnb[lane + 64]) * keep;
    o[lane + 96] = ((v3 - mu) * rs * lng[lane + 96] + lnb[lane + 96]) * keep;
  }
}

extern "C" void kernel_launch(void* const* d_in, const int* in_sizes, int n_in,
                              void* d_out, int out_size, void* d_ws, size_t ws_size,
                              hipStream_t stream) {
  (void)in_sizes; (void)n_in; (void)d_ws; (void)ws_size; (void)out_size;
  const float* x       = (const float*)d_in[0];
  const float* ef      = (const float*)d_in[1];
  const float* ap      = (const float*)d_in[2];
  const unsigned char* pm = (const unsigned char*)d_in[3];
  const float* Wq      = (const float*)d_in[4];
  const float* Wk      = (const float*)d_in[5];
  const float* Wv      = (const float*)d_in[6];
  const float* W1      = (const float*)d_in[7];
  const float* b1      = (const float*)d_in[8];
  const float* W2      = (const float*)d_in[9];
  const float* b2      = (const float*)d_in[10];
  const float* Wfuse   = (const float*)d_in[11];
  const float* Wtheta  = (const float*)d_in[12];
  const float* lng     = (const float*)d_in[13];
  const float* lnb     = (const float*)d_in[14];
  const float* physw   = (const float*)d_in[15];
  const float* priorw  = (const float*)d_in[16];
  float* out = (float*)d_out;

  fused_spatial_attn<<<BT, 256, 0, stream>>>(x, ef, ap, pm, Wq, Wk, Wv, W1, b1,
                                             W2, b2, Wfuse, Wtheta, lng, lnb,
                                             physw, priorw, out);
}